// LayerSelector_77086073028805
// MI455X (gfx1250) — compile-verified
//
#include <hip/hip_runtime.h>
#include <hip/hip_bf16.h>

// LayerSelector on gfx1250:
//   selected[b] (24 x 4096) = M_b (24x32 banded: M[k,l]=probs[l-k], 0<=l-k<=8)
//                             x  F_b (32 x 4096),  via V_WMMA_F32_16X16X4_F32.
// Bandwidth-bound: ~29 MB total HBM traffic -> ~1.26 us floor at 23.3 TB/s.
// A-fragments are prebuilt per batch in d_ws in WMMA register layout so the hot
// kernel has no band/shuffle VALU work; each wave computes 4 column tiles (64
// cols) to amortize A loads and setup.

#define B_SZ        32
#define NLAYERS     32
#define SELK        24
#define NCH         9
#define DIM         4096
#define SEL_ELEMS   (B_SZ * SELK * DIM)   // selected_features elements in d_out
#define NTILE       4                     // 16-wide col tiles per wave

typedef float v2f __attribute__((ext_vector_type(2)));
typedef float v8f __attribute__((ext_vector_type(8)));

// ---------------- Kernel 1: probs + prebuilt WMMA A-fragments ----------------
// Grid = 32 blocks (one per batch) x 64 threads. Every thread redundantly
// computes the 9 Gumbel-softmax-hard probs (cheap), then thread t = {tile,lane}
// writes the 16 A-operand floats for its lane of row-tile `tile`:
//   A[row][k] pairs for k-steps kk=0..7, row = (lane&15)+16*tile,
//   k = 4*kk + 2*(lane>=16) (+0/+1)  -- exactly the 16x4 f32 A register layout.
__global__ void ls_probs_kernel(const float* __restrict__ logits,
                                const float* __restrict__ noise,
                                float* __restrict__ Asw,
                                float* __restrict__ out_probs) {
  const int b = blockIdx.x;
  const int t = threadIdx.x;   // 0..63

  float z[NCH], soft[NCH], p[NCH];
  float zmax = -3.402823466e38f;
#pragma unroll
  for (int c = 0; c < NCH; ++c) {
    float u = noise[b * NCH + c];
    float g = -logf(-logf(u + 1e-10f) + 1e-10f);   // gumbel noise
    float zz = logits[c] + g;                      // tau = 1.0
    z[c] = zz;
    zmax = fmaxf(zmax, zz);
  }
  float s = 0.f;
#pragma unroll
  for (int c = 0; c < NCH; ++c) { soft[c] = expf(z[c] - zmax); s += soft[c]; }
  int idx = 0; float best = -1.f;
#pragma unroll
  for (int c = 0; c < NCH; ++c) {
    soft[c] = soft[c] / s;
    if (soft[c] > best) { best = soft[c]; idx = c; }   // first max (jnp.argmax)
  }
#pragma unroll
  for (int c = 0; c < NCH; ++c) {
    float h = (c == idx) ? 1.0f : 0.0f;
    p[c] = (h + soft[c]) - soft[c];   // same float ordering as reference
  }

  if (t < NCH) out_probs[b * NCH + t] = p[t];

  const int tile = t >> 5;            // row-tile 0 (rows 0..15) or 1 (rows 16..31)
  const int lane = t & 31;
  const int row  = (lane & 15) + 16 * tile;
  const int koff = (lane >> 4) << 1;
  float* __restrict__ dst = Asw + (((size_t)b * 2 + tile) * 32 + lane) * 16;
#pragma unroll
  for (int kk = 0; kk < 8; ++kk) {
    int l = kk * 4 + koff;
    int c0 = l - row, c1 = l + 1 - row;
    dst[2 * kk]     = (c0 >= 0 && c0 < NCH) ? p[c0] : 0.0f;
    dst[2 * kk + 1] = (c1 >= 0 && c1 < NCH) ? p[c1] : 0.0f;
  }
}

// ---------------- Kernel 2: banded GEMM via WMMA f32 16x16x4 ----------------
// grid = (16, 32): blockIdx.y = batch; each of the 4 waves per block owns one
// 64-column group (4 x 16-wide tiles). F is streamed exactly once.
__global__ __launch_bounds__(128)
void ls_select_kernel(const float* __restrict__ feat,
                      const float* __restrict__ Asw,
                      float* __restrict__ out) {
  const int lane = threadIdx.x & 31;
  const int wave = threadIdx.x >> 5;
  const int b    = blockIdx.y;
  const int grp  = blockIdx.x * 4 + wave;   // 0..63
  const int n0   = grp * (16 * NTILE);

  const float* __restrict__ F = feat + (size_t)b * NLAYERS * DIM;
  const v2f* __restrict__ A0p =
      (const v2f*)(Asw + (((size_t)b * 2 + 0) * 32 + lane) * 16);
  const v2f* __restrict__ A1p =
      (const v2f*)(Asw + (((size_t)b * 2 + 1) * 32 + lane) * 16);

  const int m    = lane & 15;          // B/C column index within a tile
  const int koff = (lane >> 4) << 1;   // upper half-wave handles K+2, K+3

  v8f c0[NTILE] = {}, c1[NTILE] = {};

#pragma unroll
  for (int kk = 0; kk < 8; ++kk) {     // K = 32 in steps of 4
    const int l0 = kk * 4 + koff;
    const v2f a0 = A0p[kk];            // prebuilt A fragments (cached)
    const v2f a1 = A1p[kk];
#pragma unroll
    for (int t = 0; t < NTILE; ++t) {
      v2f bv;
      const size_t col = (size_t)(n0 + t * 16 + m);
      bv.x = F[(size_t)l0       * DIM + col];
      bv.y = F[(size_t)(l0 + 1) * DIM + col];
      c0[t] = __builtin_amdgcn_wmma_f32_16x16x4_f32(false, a0, false, bv,
                                                    (short)0, c0[t], false, false);
      c1[t] = __builtin_amdgcn_wmma_f32_16x16x4_f32(false, a1, false, bv,
                                                    (short)0, c1[t], false, false);
    }
  }

  // C/D layout: VGPR r -> row (r + 8*(lane>=16)), col = lane&15
  const int rowHalf = (lane >> 4) * 8;
  float* __restrict__ Ob = out + (size_t)b * SELK * DIM;
#pragma unroll
  for (int t = 0; t < NTILE; ++t) {
    const int col = n0 + t * 16 + m;
#pragma unroll
    for (int r = 0; r < 8; ++r) {
      int row0 = r + rowHalf;                 // 0..15, always valid
      Ob[(size_t)row0 * DIM + col] = c0[t][r];
      int row1 = 16 + r + rowHalf;            // 16..31, keep < 24
      if (row1 < SELK)
        Ob[(size_t)row1 * DIM + col] = c1[t][r];
    }
  }
}

extern "C" void kernel_launch(void* const* d_in, const int* in_sizes, int n_in,
                              void* d_out, int out_size, void* d_ws, size_t ws_size,
                              hipStream_t stream) {
  const float* feat   = (const float*)d_in[0];  // (32, 32, 4096) f32
  const float* logits = (const float*)d_in[1];  // (9,) f32
  const float* noise  = (const float*)d_in[2];  // (32, 9) f32
  float* out       = (float*)d_out;
  float* out_probs = out + SEL_ELEMS;
  float* Asw       = (float*)d_ws;              // 32*2*32*16 floats = 128 KB

  ls_probs_kernel<<<B_SZ, 64, 0, stream>>>(logits, noise, Asw, out_probs);

  dim3 grid(16, B_SZ);   // 64 column groups x 32 batches, 4 waves/block
  ls_select_kernel<<<grid, 128, 0, stream>>>(feat, Asw, out);
}